// BoltzmannGateSTE_36524401885762
// MI455X (gfx1250) — compile-verified
//
#include <hip/hip_runtime.h>
#include <math.h>
#include <stdint.h>

#define HIST0_BINS 2048   // bits [30:20]
#define HISTR_BINS 1024   // 10-bit refinement levels
#define BLK 256

__device__ __forceinline__ unsigned abs_bits(float f) {
  return __float_as_uint(fabsf(f));   // bit31 cleared; order-preserving for >=0
}

// ---------------------------------------------------------------------------
// Parallel top-k bucket selection, run by ONE block (the last to finish).
// ghist has NBINS bins; finds largest bin b with count(bins > b) < k <=
// count(bins >= b). ORs b<<shift into *prefix_io, writes remaining rank.
// All global loads are parallel; the only serial part is an 8-step LDS scan
// and a <=8-entry register walk.
// ---------------------------------------------------------------------------
template <int NBINS>
__device__ __forceinline__ void block_select(const unsigned* __restrict__ ghist,
                                             unsigned k,
                                             unsigned* __restrict__ prefix_io,
                                             int shift,
                                             unsigned* __restrict__ krem_out,
                                             unsigned* scratch /* LDS >= BLK */) {
  constexpr int CH = NBINS / BLK;     // bins per thread (8 or 4)
  const int t = threadIdx.x;

  unsigned bins[CH];
  unsigned mySum = 0;
  for (int j = 0; j < CH; ++j) {
    bins[j] = ghist[t * CH + j];
    mySum += bins[j];
  }
  scratch[t] = mySum;
  __syncthreads();

  // Inclusive suffix scan over thread chunks: val = sum of chunks >= t.
  unsigned val = mySum;
  for (int off = 1; off < BLK; off <<= 1) {
    unsigned other = (t + off < BLK) ? scratch[t + off] : 0u;
    __syncthreads();
    val += other;
    scratch[t] = val;
    __syncthreads();
  }
  const unsigned suffIncl = val;            // count in chunks >= t
  const unsigned suffExcl = suffIncl - mySum;  // count in chunks >  t

  if (suffExcl < k && k <= suffIncl) {      // exactly one thread matches
    unsigned cum = suffExcl;                // count strictly above current bin
    for (int j = CH - 1; j >= 0; --j) {
      unsigned c = cum + bins[j];
      if (c >= k) {
        *prefix_io |= ((unsigned)(t * CH + j)) << shift;
        *krem_out = k - cum;
        break;
      }
      cum = c;
    }
  }
}

// Last-block detection: returns true in exactly one block after all blocks
// have flushed their histograms.
__device__ __forceinline__ bool last_block_done(unsigned* done) {
  __shared__ unsigned shLast;
  __threadfence();                          // make this block's flush visible
  if (threadIdx.x == 0)
    shLast = (atomicAdd(done, 1u) == gridDim.x - 1) ? 1u : 0u;
  __syncthreads();
  if (shLast) {
    __threadfence();                        // acquire: see all other flushes
    return true;
  }
  return false;
}

// ---------------------------------------------------------------------------
// Pass 1: 11-bit histogram of bits[30:20] of |x|, streamed from HBM via CDNA5
// async global->LDS b128 copies, double-buffered with s_wait_asynccnt, counted
// with LDS ds_add_u32 atomics. Last block runs the level-0 selection.
// ---------------------------------------------------------------------------
__global__ __launch_bounds__(BLK) void hist0_kernel(const float* __restrict__ x,
                                                    unsigned n, unsigned k,
                                                    unsigned* __restrict__ ghist,
                                                    unsigned* __restrict__ prefix,
                                                    unsigned* __restrict__ krem_out,
                                                    unsigned* __restrict__ done) {
  __shared__ unsigned shHist[HIST0_BINS];
  __shared__ float4 shBuf[2][BLK];

  for (unsigned i = threadIdx.x; i < HIST0_BINS; i += BLK) shHist[i] = 0u;
  __syncthreads();

  const unsigned n4 = n >> 2;
  const unsigned numTiles = (n4 + BLK - 1) / BLK;
  const unsigned tid = threadIdx.x;

  const unsigned ldsOff0 = (unsigned)(uintptr_t)(&shBuf[0][tid]);
  const unsigned ldsOff1 = (unsigned)(uintptr_t)(&shBuf[1][tid]);
  const float4* __restrict__ x4 = (const float4*)x;

  unsigned curTile = blockIdx.x;
  int buf = 0;
  if (curTile < numTiles) {
    unsigned idx = curTile * BLK + tid;
    if (idx >= n4) idx = n4 - 1;            // clamp load; count guarded below
    unsigned long long ga = (unsigned long long)(uintptr_t)(x4 + idx);
    asm volatile("global_load_async_to_lds_b128 %0, %1, off"
                 :: "v"(ldsOff0), "v"(ga) : "memory");
  }
  while (curTile < numTiles) {
    unsigned nextTile = curTile + gridDim.x;
    if (nextTile < numTiles) {
      unsigned idx = nextTile * BLK + tid;
      if (idx >= n4) idx = n4 - 1;
      unsigned long long ga = (unsigned long long)(uintptr_t)(x4 + idx);
      unsigned dst = (buf == 0) ? ldsOff1 : ldsOff0;
      asm volatile("global_load_async_to_lds_b128 %0, %1, off"
                   :: "v"(dst), "v"(ga) : "memory");
      asm volatile("s_wait_asynccnt 1" ::: "memory");  // cur done, prefetch in flight
    } else {
      asm volatile("s_wait_asynccnt 0" ::: "memory");
    }
    float4 v = shBuf[buf][tid];             // ds_load_b128 of this lane's slot
    unsigned idx = curTile * BLK + tid;
    if (idx < n4) {
      atomicAdd(&shHist[abs_bits(v.x) >> 20], 1u);
      atomicAdd(&shHist[abs_bits(v.y) >> 20], 1u);
      atomicAdd(&shHist[abs_bits(v.z) >> 20], 1u);
      atomicAdd(&shHist[abs_bits(v.w) >> 20], 1u);
    }
    buf ^= 1;
    curTile = nextTile;
  }
  // scalar tail if n % 4 != 0
  for (unsigned i = (n4 << 2) + blockIdx.x * BLK + tid; i < n; i += gridDim.x * BLK) {
    atomicAdd(&shHist[abs_bits(x[i]) >> 20], 1u);
  }

  __syncthreads();
  for (unsigned i = threadIdx.x; i < HIST0_BINS; i += BLK) {
    unsigned c = shHist[i];
    if (c) atomicAdd(&ghist[i], c);
  }

  if (last_block_done(done)) {
    block_select<HIST0_BINS>(ghist, k, prefix, 20, krem_out, shHist);
  }
}

// ---------------------------------------------------------------------------
// Refinement histogram (L2-resident sweep): count elements whose high bits
// match the selected prefix, binned on the next 10 bits. Last block selects.
// ---------------------------------------------------------------------------
__global__ __launch_bounds__(BLK) void hist_refine_kernel(const float* __restrict__ x,
                                                          unsigned n,
                                                          int match_shift, int bin_shift,
                                                          unsigned* __restrict__ ghist,
                                                          const unsigned* __restrict__ krem_in,
                                                          unsigned* __restrict__ prefix,
                                                          unsigned* __restrict__ krem_out,
                                                          unsigned* __restrict__ done) {
  __shared__ unsigned shHist[HISTR_BINS];
  for (unsigned i = threadIdx.x; i < HISTR_BINS; i += BLK) shHist[i] = 0u;
  __syncthreads();

  const unsigned prefHi = (*prefix) >> match_shift;
  const unsigned n4 = n >> 2;
  const float4* __restrict__ x4 = (const float4*)x;
  const unsigned stride = gridDim.x * BLK;

  for (unsigned i = blockIdx.x * BLK + threadIdx.x; i < n4; i += stride) {
    float4 v = x4[i];
    unsigned u;
    u = abs_bits(v.x); if ((u >> match_shift) == prefHi) atomicAdd(&shHist[(u >> bin_shift) & (HISTR_BINS - 1)], 1u);
    u = abs_bits(v.y); if ((u >> match_shift) == prefHi) atomicAdd(&shHist[(u >> bin_shift) & (HISTR_BINS - 1)], 1u);
    u = abs_bits(v.z); if ((u >> match_shift) == prefHi) atomicAdd(&shHist[(u >> bin_shift) & (HISTR_BINS - 1)], 1u);
    u = abs_bits(v.w); if ((u >> match_shift) == prefHi) atomicAdd(&shHist[(u >> bin_shift) & (HISTR_BINS - 1)], 1u);
  }
  for (unsigned i = (n4 << 2) + blockIdx.x * BLK + threadIdx.x; i < n; i += stride) {
    unsigned u = abs_bits(x[i]);
    if ((u >> match_shift) == prefHi) atomicAdd(&shHist[(u >> bin_shift) & (HISTR_BINS - 1)], 1u);
  }

  __syncthreads();
  for (unsigned i = threadIdx.x; i < HISTR_BINS; i += BLK) {
    unsigned c = shHist[i];
    if (c) atomicAdd(&ghist[i], c);
  }

  if (last_block_done(done)) {
    block_select<HISTR_BINS>(ghist, *krem_in, prefix, bin_shift, krem_out, shHist);
  }
}

// ---------------------------------------------------------------------------
// Final mask: out = (bits(|x|) >= T) ? x : 0.  x read from L2, out to HBM.
// ---------------------------------------------------------------------------
__global__ __launch_bounds__(BLK) void mask_kernel(const float* __restrict__ x,
                                                   float* __restrict__ out,
                                                   const unsigned* __restrict__ thr_p,
                                                   unsigned n) {
  const unsigned T = *thr_p;
  const unsigned n4 = n >> 2;
  const float4* __restrict__ x4 = (const float4*)x;
  float4* __restrict__ o4 = (float4*)out;
  const unsigned stride = gridDim.x * BLK;

  for (unsigned i = blockIdx.x * BLK + threadIdx.x; i < n4; i += stride) {
    float4 v = x4[i];
    float4 r;
    r.x = (abs_bits(v.x) >= T) ? v.x : 0.0f;
    r.y = (abs_bits(v.y) >= T) ? v.y : 0.0f;
    r.z = (abs_bits(v.z) >= T) ? v.z : 0.0f;
    r.w = (abs_bits(v.w) >= T) ? v.w : 0.0f;
    o4[i] = r;
  }
  for (unsigned i = (n4 << 2) + blockIdx.x * BLK + threadIdx.x; i < n; i += stride) {
    float f = x[i];
    out[i] = (abs_bits(f) >= T) ? f : 0.0f;
  }
}

extern "C" void kernel_launch(void* const* d_in, const int* in_sizes, int n_in,
                              void* d_out, int out_size, void* d_ws, size_t ws_size,
                              hipStream_t stream) {
  const float* x = (const float*)d_in[0];
  float* out = (float*)d_out;
  const unsigned n = (unsigned)in_sizes[0];

  unsigned* ws     = (unsigned*)d_ws;
  unsigned* hist0  = ws;                       // 2048
  unsigned* hist1  = hist0 + HIST0_BINS;       // 1024
  unsigned* hist2  = hist1 + HISTR_BINS;       // 1024
  unsigned* prefix = hist2 + HISTR_BINS;       // [0] = prefix, becomes threshold
  unsigned* krem1  = prefix + 1;
  unsigned* krem2  = prefix + 2;
  unsigned* krem3  = prefix + 3;
  unsigned* done0  = prefix + 4;
  unsigned* done1  = prefix + 5;
  unsigned* done2  = prefix + 6;

  const size_t zero_bytes = (HIST0_BINS + 2 * HISTR_BINS + 8) * sizeof(unsigned);
  hipMemsetAsync(d_ws, 0, zero_bytes, stream);

  // k exactly as the Python reference: int(n * (1/e)), truncation in double.
  const double FRACTION = 1.0 / M_E;
  unsigned long long k64 = (unsigned long long)((double)n * FRACTION);
  if (k64 < 1) k64 = 1;
  if (k64 >= (unsigned long long)n) {          // reference early-out: pass-through
    hipMemcpyAsync(out, x, (size_t)n * sizeof(float), hipMemcpyDeviceToDevice, stream);
    return;
  }
  const unsigned k = (unsigned)k64;

  const int histBlocks = 2048;                 // pass 1: HBM stream via async->LDS
  const int refBlocks  = 4096;                 // refinement/mask: L2-resident sweeps

  hist0_kernel<<<histBlocks, BLK, 0, stream>>>(x, n, k, hist0, prefix, krem1, done0);
  hist_refine_kernel<<<refBlocks, BLK, 0, stream>>>(x, n, 20, 10, hist1, krem1,
                                                    prefix, krem2, done1);
  hist_refine_kernel<<<refBlocks, BLK, 0, stream>>>(x, n, 10, 0, hist2, krem2,
                                                    prefix, krem3, done2);
  mask_kernel<<<refBlocks, BLK, 0, stream>>>(x, out, prefix, n);
}